// LengthRegulator_19121194402329
// MI455X (gfx1250) — compile-verified
//
#include <hip/hip_runtime.h>
#include <math.h>

typedef __attribute__((ext_vector_type(2))) float v2f;
typedef __attribute__((ext_vector_type(8))) float v8f;

#define BB 16
#define LL 512
#define FF 384
#define KK 3
#define MAXOUT 4096
#define NPOS (BB*LL)
#define MTB 32                 // M rows (positions) per block
#define SYP (FF + 4)           // padded LN-row stride

// ---------------------------------------------------------------------------
// Weight transform: w[Cout][Cin][K] -> fragment-ready float2 layout.
// Logical GEMM B matrix: W[r][c], r = k*FF + f (r = 0..1151), c = output ch.
// Stored as pairs: wt2[kp][c] = { W[2kp][c], W[2kp+1][c] }, kp = 0..575.
// A lane's B fragment for 4-K block q, half kh is ONE b64 load at kp = 2q+kh,
// lane-consecutive in c (fully coalesced, L2-resident: 1.77 MB << 192 MB L2).
// ---------------------------------------------------------------------------
__global__ __launch_bounds__(256)
void wtrans_kernel(const float* __restrict__ w, float* __restrict__ wt2) {
  int o = blockIdx.x * 256 + threadIdx.x;   // exactly (KK*FF/2)*FF = 221184
  int c  = o % FF;
  int kp = o / FF;                          // 0..575
  int r0 = kp * 2;                          // even -> f0 even, same k for r0,r0+1
  int k0 = r0 / FF, f0 = r0 - k0 * FF;
  float2 v;
  v.x = w[(c * FF + f0) * KK + k0];
  v.y = w[(c * FF + f0 + 1) * KK + k0];
  ((float2*)wt2)[(size_t)kp * FF + c] = v;
}

// ---------------------------------------------------------------------------
// Fused Conv1d(k=3, pad=1) + bias + LayerNorm + ReLU via V_WMMA_F32_16X16X4_F32.
// One block per 32 consecutive positions (same batch since 32 | 512).
// Each of 8 waves: 2 M-tiles x 3 N-tiles (6 accumulators); the 3 B fragments
// are shared by both M-tiles -> 2 ds_b64 + 3 global_b64 per 6 WMMAs.
// 8 waves x 48 channels = all 384 output channels -> LN fuses in-block.
// mode==0: write activated hidden to y.  mode==1: fuse linear head + exp -> dur.
// ---------------------------------------------------------------------------
__global__ __launch_bounds__(256)
void conv_ln_kernel(const float* __restrict__ x, const float* __restrict__ wt2,
                    const float* __restrict__ bias, const float* __restrict__ gamma,
                    const float* __restrict__ beta, const float* __restrict__ lw,
                    const float* __restrict__ lb, float* __restrict__ y,
                    float* __restrict__ dur, int mode)
{
  // Union buffer: phase 1 = input halo (34 rows x 384), phase 2 = LN staging
  // (32 rows x 388). Temporally disjoint; accumulators bridge the phases.
  __shared__ float smem[(MTB + 2) * FF];     // 13056 floats = 52.2 KB
  float* sx = smem;
  float* sy = smem;

  const int tid  = threadIdx.x;
  const int lane = tid & 31;
  const int wv   = tid >> 5;
  const int mtile = blockIdx.x;        // 0..255
  const int b  = mtile >> 4;           // 16 tiles per batch row
  const int l0 = (mtile & 15) * MTB;

  // Phase 1a: stage input halo rows l0-1 .. l0+32 into LDS (coalesced)
  for (int i = tid; i < (MTB + 2) * FF; i += 256) {
    int rr = i / FF, ff = i - rr * FF;
    int l = l0 + rr - 1;
    sx[i] = (l >= 0 && l < LL) ? x[((size_t)b * LL + l) * FF + ff] : 0.f;
  }
  __syncthreads();

  // Phase 1b: WMMA GEMM, K = 3*384 in steps of 4
  const int mrow  = lane & 15;         // M row within tile / N col (B,C)
  const int khalf = lane >> 4;         // which K pair of the 4-block
  const int kh2   = khalf << 1;
  v8f acc0a = {}, acc1a = {}, acc2a = {};   // M rows 0..15
  v8f acc0b = {}, acc1b = {}, acc2b = {};   // M rows 16..31
  const int c0 = (wv * 3 + 0) * 16 + mrow;
  const int c1 = (wv * 3 + 1) * 16 + mrow;
  const int c2 = (wv * 3 + 2) * 16 + mrow;

  // B fragment pointer: pair index kp = 2q + khalf, q advancing 1 per f-step
  const v2f* bp = (const v2f*)wt2 + (size_t)khalf * FF;

  for (int k = 0; k < KK; ++k) {
    const float* sxrowA = sx + (mrow + k) * FF;        // A rows, M-tile 0
    const float* sxrowB = sxrowA + 16 * FF;            // A rows, M-tile 1
    for (int f = 0; f < FF; f += 4) {
      v2f aA = *(const v2f*)(sxrowA + f + kh2);        // ds_load_b64 (paired)
      v2f aB = *(const v2f*)(sxrowB + f + kh2);
      v2f b0 = bp[c0];                                 // 3 coalesced b64 loads
      v2f b1 = bp[c1];                                 // shared by both M-tiles
      v2f b2 = bp[c2];
      acc0a = __builtin_amdgcn_wmma_f32_16x16x4_f32(false, aA, false, b0, (short)0, acc0a, false, false);
      acc1a = __builtin_amdgcn_wmma_f32_16x16x4_f32(false, aA, false, b1, (short)0, acc1a, false, false);
      acc2a = __builtin_amdgcn_wmma_f32_16x16x4_f32(false, aA, false, b2, (short)0, acc2a, false, false);
      acc0b = __builtin_amdgcn_wmma_f32_16x16x4_f32(false, aB, false, b0, (short)0, acc0b, false, false);
      acc1b = __builtin_amdgcn_wmma_f32_16x16x4_f32(false, aB, false, b1, (short)0, acc1b, false, false);
      acc2b = __builtin_amdgcn_wmma_f32_16x16x4_f32(false, aB, false, b2, (short)0, acc2b, false, false);
      bp += 2 * FF;                                    // next 4-K block (2 pairs)
    }
  }
  __syncthreads();   // all waves done reading sx; smem becomes LN buffer

  // Phase 2a: spill C/D (+bias): VGPR i -> M = i + 8*lane[4], N = lane[3:0]
  {
    const int mhi = khalf * 8;
    const float bb0 = bias[c0], bb1 = bias[c1], bb2 = bias[c2];
    #pragma unroll
    for (int i = 0; i < 8; ++i) {
      int m = mhi + i;
      sy[m * SYP + c0] = acc0a[i] + bb0;
      sy[m * SYP + c1] = acc1a[i] + bb1;
      sy[m * SYP + c2] = acc2a[i] + bb2;
      sy[(m + 16) * SYP + c0] = acc0b[i] + bb0;
      sy[(m + 16) * SYP + c1] = acc1b[i] + bb1;
      sy[(m + 16) * SYP + c2] = acc2b[i] + bb2;
    }
  }
  __syncthreads();

  // Phase 2b: LayerNorm + ReLU (each wave: 4 rows, shuffle reduction)
  for (int rr = 0; rr < 4; ++rr) {
    int m = wv * 4 + rr;
    const float* row = sy + m * SYP;
    float s = 0.f, sq = 0.f;
    for (int c = lane; c < FF; c += 32) { float v = row[c]; s += v; sq += v * v; }
    for (int off = 16; off; off >>= 1) { s += __shfl_xor(s, off); sq += __shfl_xor(sq, off); }
    float mu  = s * (1.f / FF);
    float var = sq * (1.f / FF) - mu * mu;
    float inv = rsqrtf(var + 1e-5f);
    size_t pos = (size_t)mtile * MTB + m;
    if (mode == 0) {
      for (int c = lane; c < FF; c += 32) {
        float v = (row[c] - mu) * inv * gamma[c] + beta[c];
        y[pos * FF + c] = fmaxf(v, 0.f);
      }
    } else {
      float dot = 0.f;
      for (int c = lane; c < FF; c += 32) {
        float v = fmaxf((row[c] - mu) * inv * gamma[c] + beta[c], 0.f);
        dot += v * lw[c];
      }
      for (int off = 16; off; off >>= 1) dot += __shfl_xor(dot, off);
      if (lane == 0) dur[pos] = expf(fmaxf(dot + lb[0], 0.f));
    }
  }
}

// ---------------------------------------------------------------------------
// Inclusive cumsum of target durations per batch row (Hillis-Steele in LDS).
// ---------------------------------------------------------------------------
__global__ __launch_bounds__(512)
void cumsum_kernel(const int* __restrict__ tgt, int* __restrict__ ends) {
  __shared__ int sd[LL];
  int b = blockIdx.x, t = threadIdx.x;
  sd[t] = tgt[b * LL + t];
  __syncthreads();
  for (int off = 1; off < LL; off <<= 1) {
    int v = sd[t];
    if (t >= off) v += sd[t - off];
    __syncthreads();
    sd[t] = v;
    __syncthreads();
  }
  ends[b * LL + t] = sd[t];
}

// ---------------------------------------------------------------------------
// Length regulator: one wave per output row. searchsorted(right) + float4 copy
// or zero fill (implements the mask). Pure streaming, ~100 MB of writes ->
// runs at the 23.3 TB/s HBM floor (~4.3 us).
// ---------------------------------------------------------------------------
__global__ __launch_bounds__(256)
void regulate_kernel(const float* __restrict__ x, const int* __restrict__ ends,
                     float* __restrict__ out) {
  int lane = threadIdx.x & 31;
  int wv   = threadIdx.x >> 5;
  int row  = blockIdx.x * 8 + wv;        // b*MAXOUT + t
  int b = row >> 12;
  int t = row & (MAXOUT - 1);
  const int* e = ends + b * LL;
  float4* dst = (float4*)(out + (size_t)row * FF);
  int total = e[LL - 1];
  if (t >= total) {
    float4 z = make_float4(0.f, 0.f, 0.f, 0.f);
    for (int i = lane; i < FF / 4; i += 32) dst[i] = z;
  } else {
    int lo = 0, hi = LL;                 // first idx with e[idx] > t
    while (lo < hi) { int mid = (lo + hi) >> 1; if (e[mid] <= t) lo = mid + 1; else hi = mid; }
    int idx = lo < LL ? lo : LL - 1;
    const float4* src = (const float4*)(x + ((size_t)b * LL + idx) * FF);
    for (int i = lane; i < FF / 4; i += 32) dst[i] = src[i];
  }
}

// ---------------------------------------------------------------------------
extern "C" void kernel_launch(void* const* d_in, const int* in_sizes, int n_in,
                              void* d_out, int out_size, void* d_ws, size_t ws_size,
                              hipStream_t stream) {
  const float* input   = (const float*)d_in[0];
  const int*   target  = (const int*)  d_in[1];
  const float* conv1_w = (const float*)d_in[2];
  const float* conv1_b = (const float*)d_in[3];
  const float* ln1_g   = (const float*)d_in[4];
  const float* ln1_b   = (const float*)d_in[5];
  const float* conv2_w = (const float*)d_in[6];
  const float* conv2_b = (const float*)d_in[7];
  const float* ln2_g   = (const float*)d_in[8];
  const float* ln2_b   = (const float*)d_in[9];
  const float* lin_w   = (const float*)d_in[10];
  const float* lin_b   = (const float*)d_in[11];

  float* out = (float*)d_out;
  float* dur = out + (size_t)BB * MAXOUT * FF;   // durations appended after main output

  char* ws = (char*)d_ws;
  float* wt2  = (float*)ws;                                              // 1.77 MB
  float* h1   = (float*)(ws + (size_t)KK * FF * FF * sizeof(float));     // 12.6 MB
  int*   ends = (int*)  (ws + (size_t)KK * FF * FF * sizeof(float)
                            + (size_t)NPOS * FF * sizeof(float));        // 32 KB

  // Layer 1: weight transform, fused conv+LN+ReLU -> h1
  wtrans_kernel<<<(KK * FF * FF / 2) / 256, 256, 0, stream>>>(conv1_w, wt2);
  conv_ln_kernel<<<NPOS / MTB, 256, 0, stream>>>(input, wt2, conv1_b, ln1_g, ln1_b,
                                                 lin_w, lin_b, h1, dur, 0);
  // Layer 2: reuse wt2 buffer (stream-ordered), fuse linear head + exp -> durations
  wtrans_kernel<<<(KK * FF * FF / 2) / 256, 256, 0, stream>>>(conv2_w, wt2);
  conv_ln_kernel<<<NPOS / MTB, 256, 0, stream>>>(h1, wt2, conv2_b, ln2_g, ln2_b,
                                                 lin_w, lin_b, h1, dur, 1);
  // Length regulation
  cumsum_kernel<<<BB, LL, 0, stream>>>(target, ends);
  regulate_kernel<<<(BB * MAXOUT) / 8, 256, 0, stream>>>(input, ends, out);
}